// XtrPairwiseCELoss_78434692760234
// MI455X (gfx1250) — compile-verified
//
#include <hip/hip_runtime.h>
#include <math.h>

typedef __attribute__((ext_vector_type(2))) float v2f;
typedef __attribute__((ext_vector_type(8))) float v8f;

#define B_  64
#define N_  32
#define S_  512
#define D_  128
#define NEG_BIG -3.402823466e38f

// Insert (val,row) into a sorted top-3 list (v1>=v2>=v3).
__device__ __forceinline__ void top3_insert(float v, int r,
                                            float& v1, int& r1,
                                            float& v2, int& r2,
                                            float& v3, int& r3) {
    if (v > v1)      { v3 = v2; r3 = r2; v2 = v1; r2 = r1; v1 = v; r1 = r; }
    else if (v > v2) { v3 = v2; r3 = r2; v2 = v;  r2 = r; }
    else if (v > v3) { v3 = v;  r3 = r; }
}

// One block per (b = blockIdx.y query batch, c = blockIdx.x doc batch).
// Computes the 32x512 score tile Q_b (32x128) * Dc^T (128x512) with
// v_wmma_f32_16x16x4_f32, fuses global top-3 (value + query-row index),
// and writes the scalar XTR score for this pair.
__global__ __launch_bounds__(256)
void xtr_pair_score_kernel(const float* __restrict__ q,
                           const float* __restrict__ d,
                           float* __restrict__ scores) {
    const int c    = blockIdx.x;
    const int b    = blockIdx.y;
    const int tid  = threadIdx.x;
    const int lane = tid & 31;
    const int wave = tid >> 5;
    const int half = lane >> 4;   // 0: lanes 0-15, 1: lanes 16-31
    const int l16  = lane & 15;

    const float* __restrict__ Q  = q + (size_t)b * N_ * D_;
    const float* __restrict__ Dc = d + (size_t)c * S_ * D_;

    // A-matrix f32 16x4 layout: lane L -> M = L%16, VGPR v -> K = v + 2*(L/16)
    //   => each lane needs the contiguous pair Q[m][k0 + 2*half .. +1]
    // B-matrix f32 4x16 layout: lane L -> N = L%16, VGPR v -> K = v + 2*(L/16)
    //   => each lane needs the contiguous pair Dc[s][k0 + 2*half .. +1]
    const float* aRow0 = Q + (size_t)(l16 +  0) * D_ + 2 * half;  // M-tile 0
    const float* aRow1 = Q + (size_t)(l16 + 16) * D_ + 2 * half;  // M-tile 1
    const float* bRow0 = Dc + (size_t)(wave * 64 +  0 + l16) * D_ + 2 * half;
    const float* bRow1 = Dc + (size_t)(wave * 64 + 16 + l16) * D_ + 2 * half;
    const float* bRow2 = Dc + (size_t)(wave * 64 + 32 + l16) * D_ + 2 * half;
    const float* bRow3 = Dc + (size_t)(wave * 64 + 48 + l16) * D_ + 2 * half;

    v8f acc[2][4] = {};   // 2 M-tiles x 4 N-tiles of 16x16 f32

    #pragma unroll 4
    for (int ks = 0; ks < D_ / 4; ++ks) {
        const int ko = ks * 4;
        v2f a0 = *(const v2f*)(aRow0 + ko);
        v2f a1 = *(const v2f*)(aRow1 + ko);
        v2f b0 = *(const v2f*)(bRow0 + ko);
        v2f b1 = *(const v2f*)(bRow1 + ko);
        v2f b2 = *(const v2f*)(bRow2 + ko);
        v2f b3 = *(const v2f*)(bRow3 + ko);
        // (neg_a, A, neg_b, B, c_mod, C, reuse_a, reuse_b)
        acc[0][0] = __builtin_amdgcn_wmma_f32_16x16x4_f32(false, a0, false, b0, (short)0, acc[0][0], false, false);
        acc[1][0] = __builtin_amdgcn_wmma_f32_16x16x4_f32(false, a1, false, b0, (short)0, acc[1][0], false, false);
        acc[0][1] = __builtin_amdgcn_wmma_f32_16x16x4_f32(false, a0, false, b1, (short)0, acc[0][1], false, false);
        acc[1][1] = __builtin_amdgcn_wmma_f32_16x16x4_f32(false, a1, false, b1, (short)0, acc[1][1], false, false);
        acc[0][2] = __builtin_amdgcn_wmma_f32_16x16x4_f32(false, a0, false, b2, (short)0, acc[0][2], false, false);
        acc[1][2] = __builtin_amdgcn_wmma_f32_16x16x4_f32(false, a1, false, b2, (short)0, acc[1][2], false, false);
        acc[0][3] = __builtin_amdgcn_wmma_f32_16x16x4_f32(false, a0, false, b3, (short)0, acc[0][3], false, false);
        acc[1][3] = __builtin_amdgcn_wmma_f32_16x16x4_f32(false, a1, false, b3, (short)0, acc[1][3], false, false);
    }

    // Per-lane top-3 over this lane's 64 accumulator elements.
    // C/D layout: lane L, VGPR v -> N = L%16, M = v + 8*(L/16); query row = M + 16*mt.
    float v1 = NEG_BIG, v2 = NEG_BIG, v3 = NEG_BIG;
    int   r1 = 0, r2 = 0, r3 = 0;
    #pragma unroll
    for (int mt = 0; mt < 2; ++mt)
        #pragma unroll
        for (int nt = 0; nt < 4; ++nt)
            #pragma unroll
            for (int e = 0; e < 8; ++e) {
                float val = acc[mt][nt][e];
                int   row = e + 8 * half + 16 * mt;
                top3_insert(val, row, v1, r1, v2, r2, v3, r3);
            }

    // Wave-level top-3 merge (wave32 butterfly).
    #pragma unroll
    for (int off = 16; off >= 1; off >>= 1) {
        float ov1 = __shfl_xor(v1, off, 32); int or1 = __shfl_xor(r1, off, 32);
        float ov2 = __shfl_xor(v2, off, 32); int or2 = __shfl_xor(r2, off, 32);
        float ov3 = __shfl_xor(v3, off, 32); int or3 = __shfl_xor(r3, off, 32);
        top3_insert(ov1, or1, v1, r1, v2, r2, v3, r3);
        top3_insert(ov2, or2, v1, r1, v2, r2, v3, r3);
        top3_insert(ov3, or3, v1, r1, v2, r2, v3, r3);
    }

    // Block-level merge across 8 waves via LDS.
    __shared__ float sv[24];
    __shared__ int   sr[24];
    if (lane == 0) {
        sv[wave * 3 + 0] = v1; sr[wave * 3 + 0] = r1;
        sv[wave * 3 + 1] = v2; sr[wave * 3 + 1] = r2;
        sv[wave * 3 + 2] = v3; sr[wave * 3 + 2] = r3;
    }
    __syncthreads();
    if (tid == 0) {
        float m1 = NEG_BIG, m2 = NEG_BIG, m3 = NEG_BIG;
        int   q1 = 0, q2 = 0, q3 = 0;
        #pragma unroll
        for (int i = 0; i < 24; ++i)
            top3_insert(sv[i], sr[i], m1, q1, m2, q2, m3, q3);
        // score = sum over rows-with-hits of per-row max / (#distinct rows)
        float sum = m1;
        float z   = 1.0f;
        if (q2 != q1)              { sum += m2; z += 1.0f; }
        if (q3 != q1 && q3 != q2)  { sum += m3; z += 1.0f; }
        scores[b * B_ + c] = sum / z;
    }
}

// pos = diag(scores); neg = rowwise max with diag suppressed by -1e6;
// loss = mean(softplus(neg - pos))
__global__ __launch_bounds__(64)
void xtr_loss_kernel(const float* __restrict__ scores, float* __restrict__ out) {
    __shared__ float red[B_];
    const int bb = threadIdx.x;
    const float* row = scores + bb * B_;
    float pos = row[bb];
    float neg = NEG_BIG;
    #pragma unroll 8
    for (int cc = 0; cc < B_; ++cc) {
        float s = row[cc] - (cc == bb ? 1.0e6f : 0.0f);
        neg = fmaxf(neg, s);
    }
    float x  = neg - pos;
    float sp = fmaxf(x, 0.0f) + log1pf(expf(-fabsf(x)));  // stable softplus
    red[bb] = sp;
    __syncthreads();
    if (bb == 0) {
        float s = 0.0f;
        #pragma unroll
        for (int i = 0; i < B_; ++i) s += red[i];
        out[0] = s * (1.0f / (float)B_);
    }
}

extern "C" void kernel_launch(void* const* d_in, const int* in_sizes, int n_in,
                              void* d_out, int out_size, void* d_ws, size_t ws_size,
                              hipStream_t stream) {
    (void)in_sizes; (void)n_in; (void)out_size; (void)ws_size;
    const float* q = (const float*)d_in[0];   // (64, 32, 128) f32
    const float* d = (const float*)d_in[1];   // (64, 512, 128) f32
    float* scores  = (float*)d_ws;            // (64, 64) f32 scratch

    dim3 grid(B_, B_);                        // blockIdx.x = c, blockIdx.y = b
    xtr_pair_score_kernel<<<grid, 256, 0, stream>>>(q, d, scores);
    xtr_loss_kernel<<<1, 64, 0, stream>>>(scores, (float*)d_out);
}